// MixtureOfExpertsLoss_88029649699669
// MI455X (gfx1250) — compile-verified
//
#include <hip/hip_runtime.h>
#include <float.h>
#include <stdint.h>

// ---------------------------------------------------------------------------
// MoE loss: base CE (ignore_index=0) + 0.01*var(counts) + 0.01*var(gate load)
// B=4 S=2048 V=32000 E=8 K=2.  Memory-bound: 1.05 GB of logits -> ~45us floor
// at 23.3 TB/s.  Single-pass online softmax streamed via gfx1250 async
// VMEM->LDS (ASYNCcnt) with a 4-deep per-wave pipeline.
// ---------------------------------------------------------------------------

#define V_      32000
#define V4_     (V_ / 4)      // 8000 float4 per row
#define ROWS_   8192          // B*S
#define E_      8
#define AUX_W   0.01f
#define LB_W    0.01f

// ---- gfx1250 async VMEM->LDS builtins (probe-confirmed present) -----------
#if __has_builtin(__builtin_amdgcn_global_load_async_to_lds_b128) && \
    __has_builtin(__builtin_amdgcn_s_wait_asynccnt)
#define HAVE_ASYNC_LDS 1
#else
#define HAVE_ASYNC_LDS 0
#warning "CDNA5 probe: async-to-LDS builtins NOT declared -> direct-load path"
#endif

#if HAVE_ASYNC_LDS
// Builtin params: v4i (gcc vector_size 16) pointers -- global src AS(1),
// LDS dest AS(3) -- then imm offset, imm cpol.
typedef int v4i_ __attribute__((vector_size(16)));
typedef __attribute__((address_space(1))) v4i_ as1_v4i;
typedef __attribute__((address_space(3))) v4i_ as3_v4i;
__device__ static inline void async_cp16(const void* g, void* l) {
    __builtin_amdgcn_global_load_async_to_lds_b128(
        (as1_v4i*)const_cast<void*>(g), (as3_v4i*)l, 0, 0);
}
#endif

// online-softmax update with a float4
#define ONLINE4(xv)                                                          \
    do {                                                                     \
        float mx_ = fmaxf(fmaxf((xv).x, (xv).y), fmaxf((xv).z, (xv).w));     \
        float nm_ = fmaxf(m, mx_);                                           \
        s = s * __expf(m - nm_) + __expf((xv).x - nm_) + __expf((xv).y - nm_)\
            + __expf((xv).z - nm_) + __expf((xv).w - nm_);                   \
        m = nm_;                                                             \
    } while (0)

// ---------------------------------------------------------------------------
// one block (8 wave32) per row of 32000 logits; single pass online softmax.
// Placed first in the file so the disasm snippet shows the async inner loop.
__global__ __launch_bounds__(256) void moe_ce(const float* __restrict__ logits,
                                              const int* __restrict__ labels,
                                              float* __restrict__ ws) {
    const int row  = blockIdx.x;
    const int tid  = threadIdx.x;
    const int wave = tid >> 5;
    const int lane = tid & 31;
    const float4* row4 = (const float4*)(logits + (size_t)row * V_);

    float m = -FLT_MAX, s = 0.0f;

#if HAVE_ASYNC_LDS
    // Each wave streams its 1000-float4 slice through a private 4-deep LDS
    // ring (2 KB in flight per wave; 16 KB total of the WGP's 320 KB).
    __shared__ float4 stage[4][8][32];
    const int base = wave * 1000;                 // slice start (float4 units)
    // prologue: issue tiles 0..2 (ASYNCcnt = 3)
#pragma unroll
    for (int k = 0; k < 3; ++k)
        async_cp16(row4 + base + k * 32 + lane, &stage[k][wave][lane]);
    // steady state: keep 4 tiles outstanding (tiles 0..27 consumed here)
    for (int i = 0; i < 28; ++i) {
        async_cp16(row4 + base + (i + 3) * 32 + lane, &stage[(i + 3) & 3][wave][lane]);
        __builtin_amdgcn_s_wait_asynccnt(3);      // tile i has landed
        float4 x = stage[i & 3][wave][lane];
        ONLINE4(x);
    }
    // drain: immediates required, so explicit epilogue (tiles 28,29,30)
    __builtin_amdgcn_s_wait_asynccnt(2); { float4 x = stage[0][wave][lane]; ONLINE4(x); }
    __builtin_amdgcn_s_wait_asynccnt(1); { float4 x = stage[1][wave][lane]; ONLINE4(x); }
    __builtin_amdgcn_s_wait_asynccnt(0); { float4 x = stage[2][wave][lane]; ONLINE4(x); }
    // slice remainder: 1000 = 31*32 + 8 -> lanes 0..7 take the tail directly
    if (lane < 8) {
        float4 x = row4[base + 992 + lane];
        ONLINE4(x);
    }
#else
    // direct 128-bit streaming loads + gfx1250 prefetch ~32KB ahead
    for (int c = tid; c < V4_; c += 256) {
        if (c + 2048 < V4_) __builtin_prefetch(row4 + c + 2048, 0, 0);
        float4 x = row4[c];
        ONLINE4(x);
    }
#endif

    // wave32 reduction of (m, s) pairs
    for (int off = 16; off > 0; off >>= 1) {
        float om = __shfl_xor(m, off, 32);
        float os = __shfl_xor(s, off, 32);
        float nm = fmaxf(m, om);
        s = s * __expf(m - nm) + os * __expf(om - nm);
        m = nm;
    }
    __shared__ float wm[8], wsum[8];
    if (lane == 0) { wm[wave] = m; wsum[wave] = s; }
    __syncthreads();
    if (tid == 0) {
        float M = wm[0], S = wsum[0];
        for (int w = 1; w < 8; ++w) {
            float nm = fmaxf(M, wm[w]);
            S = S * __expf(M - nm) + wsum[w] * __expf(wm[w] - nm);
            M = nm;
        }
        int lab = labels[row];
        if (lab != 0) {   // ignore_index == 0
            float tgt = logits[(size_t)row * V_ + lab];
            float nll = M + logf(S) - tgt;
            atomicAdd(&ws[0], nll);
            atomicAdd(&ws[1], 1.0f);
        }
    }
}

// ---------------------------------------------------------------------------
__global__ void moe_init(float* ws) {
    int t = threadIdx.x;
    if (t < 18) ws[t] = 0.0f;   // [0]=nll_sum [1]=valid [2..9]=counts [10..17]=load
}

// gate softmax load sums + expert-index counts (tiny: ~0.3 MB total traffic)
__global__ __launch_bounds__(256) void moe_gate(const float* __restrict__ gate,
                                                const int* __restrict__ eidx,
                                                float* __restrict__ ws) {
    __shared__ float lload[E_];
    __shared__ float lcnt[E_];
    const int t = threadIdx.x;
    if (t < E_) { lload[t] = 0.0f; lcnt[t] = 0.0f; }
    __syncthreads();

    const int tok = blockIdx.x * 256 + t;   // grid covers 8192 tokens exactly
    // softmax over 8 gate logits
    const float4* g4 = (const float4*)(gate + (size_t)tok * E_);
    float4 a = g4[0], b = g4[1];
    float mx = fmaxf(fmaxf(fmaxf(a.x, a.y), fmaxf(a.z, a.w)),
                     fmaxf(fmaxf(b.x, b.y), fmaxf(b.z, b.w)));
    float e0 = __expf(a.x - mx), e1 = __expf(a.y - mx);
    float e2 = __expf(a.z - mx), e3 = __expf(a.w - mx);
    float e4 = __expf(b.x - mx), e5 = __expf(b.y - mx);
    float e6 = __expf(b.z - mx), e7 = __expf(b.w - mx);
    float inv = 1.0f / (e0 + e1 + e2 + e3 + e4 + e5 + e6 + e7);
    atomicAdd(&lload[0], e0 * inv); atomicAdd(&lload[1], e1 * inv);
    atomicAdd(&lload[2], e2 * inv); atomicAdd(&lload[3], e3 * inv);
    atomicAdd(&lload[4], e4 * inv); atomicAdd(&lload[5], e5 * inv);
    atomicAdd(&lload[6], e6 * inv); atomicAdd(&lload[7], e7 * inv);

    // two expert indices per token
    int i0 = eidx[tok * 2 + 0];
    int i1 = eidx[tok * 2 + 1];
    atomicAdd(&lcnt[i0 & (E_ - 1)], 1.0f);
    atomicAdd(&lcnt[i1 & (E_ - 1)], 1.0f);

    __syncthreads();
    if (t < E_) {
        atomicAdd(&ws[2 + t],  lcnt[t]);
        atomicAdd(&ws[10 + t], lload[t]);
    }
}

__global__ void moe_final(const float* __restrict__ ws, float* __restrict__ out) {
    if (threadIdx.x == 0 && blockIdx.x == 0) {
        float base = ws[0] / fmaxf(ws[1], 1.0f);
        float cm = 0.0f, lm = 0.0f;
        for (int i = 0; i < E_; ++i) { cm += ws[2 + i]; lm += ws[10 + i]; }
        cm *= (1.0f / E_); lm *= (1.0f / E_);
        float av = 0.0f, lv = 0.0f;
        for (int i = 0; i < E_; ++i) {
            float dc = ws[2 + i] - cm;  av += dc * dc;
            float dl = ws[10 + i] - lm; lv += dl * dl;
        }
        av *= (1.0f / E_); lv *= (1.0f / E_);
        out[0] = base + AUX_W * av + LB_W * lv;
    }
}

// ---------------------------------------------------------------------------
extern "C" void kernel_launch(void* const* d_in, const int* in_sizes, int n_in,
                              void* d_out, int out_size, void* d_ws, size_t ws_size,
                              hipStream_t stream) {
    const float* logits = (const float*)d_in[0];
    const int*   labels = (const int*)d_in[1];
    const float* gate   = (const float*)d_in[2];
    const int*   eidx   = (const int*)d_in[3];
    float* ws  = (float*)d_ws;
    float* out = (float*)d_out;

    moe_init <<<1, 32, 0, stream>>>(ws);
    moe_ce   <<<ROWS_, 256, 0, stream>>>(logits, labels, ws);
    moe_gate <<<ROWS_ / 256, 256, 0, stream>>>(gate, eidx, ws);
    moe_final<<<1, 32, 0, stream>>>(ws, out);
}